// Node_feat_fusion_17712445129202
// MI455X (gfx1250) — compile-verified
//
#include <hip/hip_runtime.h>

// GCN message passing on MI455X (gfx1250):
//   out = segment_sum(hidden[src], dst); out[deg==0] = hidden[deg==0]
//
// Bound by L2 FP32 atomic throughput: hidden (12.8 MB) and out (12.8 MB) both
// fit in the 192 MB device L2, so gathers/scatters are L2-resident. Edge lists
// are streamed once (nontemporal + prefetch). No dense matmul exists in this
// op, so WMMA is intentionally not used; the CDNA5-specific paths here are
// native global_atomic_add_f32 (non-returning), b128 gathers, and
// global_prefetch_b8.

#define D_FEAT 32
#define EDGE_LANES 8   // 8 threads * float4 = one 32-float row per edge

__global__ __launch_bounds__(256) void gcn_scatter_kernel(
    const float* __restrict__ hidden,
    const int*   __restrict__ src,
    const int*   __restrict__ dst,
    float*       __restrict__ out,
    unsigned*    __restrict__ deg,
    int n_edges)
{
    long long gid = (long long)blockIdx.x * blockDim.x + threadIdx.x;
    int e = (int)(gid >> 3);      // edge index
    int t = (int)(gid & 7);       // float4 slot within the 32-float row
    if (e >= n_edges) return;

    // Streaming, read-once index data: nontemporal so it never displaces the
    // L2-resident hidden/out working set.
    int s = __builtin_nontemporal_load(src + e);
    int d = __builtin_nontemporal_load(dst + e);

    if (t == 0) {
        // Software-prefetch the edge stream well ahead (global_prefetch_b8).
        if (e + 4096 < n_edges) {
            __builtin_prefetch(src + e + 4096, 0, 0);
            __builtin_prefetch(dst + e + 4096, 0, 0);
        }
        // One in-degree atomic per edge (leader lane only).
        atomicAdd(deg + d, 1u);
    }

    // Coalesced 128-byte row gather: 8 consecutive lanes load 8 float4s of the
    // same source row -> global_load_b128, L2 hit after first touch.
    const float4 v = ((const float4*)(hidden + (long long)s * D_FEAT))[t];

    // Non-returning hardware FP32 atomics into the L2-resident accumulator.
    // unsafeAtomicAdd lowers to global_atomic_add_f32 (no CAS loop, no return
    // data -> tracked by STOREcnt only; s_endpgm's implicit wait-idle covers us).
    float* orow = out + (long long)d * D_FEAT + (t << 2);
    unsafeAtomicAdd(orow + 0, v.x);
    unsafeAtomicAdd(orow + 1, v.y);
    unsafeAtomicAdd(orow + 2, v.z);
    unsafeAtomicAdd(orow + 3, v.w);
}

__global__ __launch_bounds__(256) void gcn_finalize_kernel(
    const float*    __restrict__ hidden,
    const unsigned* __restrict__ deg,
    float*          __restrict__ out,
    int n_nodes)
{
    long long gid = (long long)blockIdx.x * blockDim.x + threadIdx.x;  // float4 id
    int n = (int)(gid >> 3);
    if (n >= n_nodes) return;
    // Nodes with no in-edges keep their old features.
    if (deg[n] == 0) {
        ((float4*)out)[gid] = ((const float4*)hidden)[gid];
    }
}

extern "C" void kernel_launch(void* const* d_in, const int* in_sizes, int n_in,
                              void* d_out, int out_size, void* d_ws, size_t ws_size,
                              hipStream_t stream)
{
    const float* hidden = (const float*)d_in[0];
    const int*   src    = (const int*)d_in[1];
    const int*   dst    = (const int*)d_in[2];
    float*       out    = (float*)d_out;

    const int n_nodes = in_sizes[0] / D_FEAT;
    const int n_edges = in_sizes[1];

    unsigned* deg = (unsigned*)d_ws;   // n_nodes u32 in scratch

    // Re-zero accumulator + degree every call (graph replays don't re-poison).
    hipMemsetAsync(out, 0, (size_t)out_size * sizeof(float), stream);
    hipMemsetAsync(deg, 0, (size_t)n_nodes * sizeof(unsigned), stream);

    {
        const int block = 256;
        long long threads = (long long)n_edges * EDGE_LANES;
        unsigned grid = (unsigned)((threads + block - 1) / block);
        gcn_scatter_kernel<<<grid, block, 0, stream>>>(hidden, src, dst, out, deg, n_edges);
    }
    {
        const int block = 256;
        long long threads = (long long)n_nodes * EDGE_LANES;  // float4 lanes
        unsigned grid = (unsigned)((threads + block - 1) / block);
        gcn_finalize_kernel<<<grid, block, 0, stream>>>(hidden, deg, out, n_nodes);
    }
}